// HierarchicalSoftmax_17308718203029
// MI455X (gfx1250) — compile-verified
//
#include <hip/hip_runtime.h>
#include <hip/hip_bf16.h>
#include <math.h>

// Problem constants (from the reference)
#define NHID    1024
#define NCLS    224     // NCLASSES
#define NPC     225     // tokens per class
#define BATCH   1024
#define CHMAX   8       // samples per class chunk staged in LDS
#define WROWS   32      // weight rows (d values) per TDM tile
#define NCHUNK  (NHID / WROWS)

typedef __attribute__((ext_vector_type(2))) float v2f;
typedef __attribute__((ext_vector_type(8))) float v8f;
typedef unsigned int u32x4 __attribute__((ext_vector_type(4)));
typedef int          i32x4 __attribute__((ext_vector_type(4)));
typedef int          i32x8 __attribute__((ext_vector_type(8)));

// ---------------------------------------------------------------------------
// Kernel A: top_logits = inputs @ top_W + top_b via V_WMMA_F32_16X16X4_F32.
// ---------------------------------------------------------------------------
__global__ void __launch_bounds__(32)
hs_top_gemm_wmma(const float* __restrict__ A,      // [BATCH, NHID]
                 const float* __restrict__ Wt,     // [NHID, NCLS]
                 const float* __restrict__ bias,   // [NCLS]
                 float* __restrict__ outLogits)    // [BATCH, NCLS]
{
    const int m0   = blockIdx.x * 16;
    const int n0   = blockIdx.y * 16;
    const int lane = threadIdx.x;
    const int half = lane >> 4;
    const int l    = lane & 15;

    const float* arow = A + (size_t)(m0 + l) * NHID;

    v8f c = {};
    for (int k = 0; k < NHID; k += 4) {
        const int ka = k + half * 2;
        v2f a;
        a.x = arow[ka + 0];
        a.y = arow[ka + 1];
        v2f b;
        b.x = Wt[(size_t)(ka + 0) * NCLS + n0 + l];
        b.y = Wt[(size_t)(ka + 1) * NCLS + n0 + l];
        c = __builtin_amdgcn_wmma_f32_16x16x4_f32(
                false, a, false, b, (short)0, c, false, false);
    }

    const int   col = n0 + l;
    const float bv  = bias[col];
#pragma unroll
    for (int r = 0; r < 8; ++r) {
        const int row = m0 + r + half * 8;
        outLogits[(size_t)row * NCLS + col] = c[r] + bv;
    }
}

// ---------------------------------------------------------------------------
// Grouping kernels (atomic list order never affects per-sample math values).
// ---------------------------------------------------------------------------
__global__ void __launch_bounds__(256)
hs_zero_counts(int* __restrict__ counts)
{
    if (threadIdx.x < NCLS) counts[threadIdx.x] = 0;
}

__global__ void __launch_bounds__(256)
hs_build_lists(const int* __restrict__ labels,
               int* __restrict__ counts,
               int* __restrict__ lists)            // [NCLS][BATCH]
{
    const int b = blockIdx.x * 256 + threadIdx.x;
    if (b < BATCH) {
        const int ct = labels[b] / NPC;
        const int p  = atomicAdd(&counts[ct], 1);
        lists[ct * BATCH + p] = b;
    }
}

// ---------------------------------------------------------------------------
// TDM issue: DMA one [WROWS x NPC] f32 weight tile into LDS.
// D# group0: count=1 | lds_addr | 57-bit global addr | type=2.
// D# group1: data_size=4B, tensor_dim0=225, tensor_dim1=1024,
//            tile_dim0=225, tile_dim1=WROWS, tensor_dim0_stride=225.
// ---------------------------------------------------------------------------
__device__ __forceinline__ void tdm_load_w_tile(const float* gsrc, unsigned lds_addr)
{
    const unsigned long long ga = (unsigned long long)(uintptr_t)gsrc;
    u32x4 g0;
    g0[0] = 1u;                                            // count=1
    g0[1] = lds_addr;                                      // LDS byte address
    g0[2] = (unsigned)ga;                                  // global addr [31:0]
    g0[3] = (unsigned)((ga >> 32) & 0x01FFFFFFu) | (2u << 30); // addr [56:32] | type=2
    i32x8 g1;
    g1[0] = (int)(2u << 16);          // data_size = 2 (4 bytes)
    g1[1] = (int)((unsigned)NPC << 16);   // tensor_dim0 lo16 (bits 63:48)
    g1[2] = (int)((unsigned)NHID << 16);  // tensor_dim0 hi=0 | tensor_dim1 lo16
    g1[3] = (int)((unsigned)NPC << 16);   // tensor_dim1 hi=0 | tile_dim0=225
    g1[4] = (int)WROWS;               // tile_dim1=WROWS | tile_dim2=0
    g1[5] = (int)NPC;                 // tensor_dim0_stride lo32
    g1[6] = 0;
    g1[7] = 0;
    i32x4 g2 = {};
    i32x4 g3 = {};
#if defined(__clang_major__) && (__clang_major__ >= 23)
    i32x8 g4 = {};
    __builtin_amdgcn_tensor_load_to_lds(g0, g1, g2, g3, g4, 0);
#else
    __builtin_amdgcn_tensor_load_to_lds(g0, g1, g2, g3, 0);
#endif
}

// ---------------------------------------------------------------------------
// Kernel G: one block per class. The class's [1024 x 225] slab is streamed
// once per chunk of up to CHMAX samples (~205 MB per full pass). Weights arrive via double-
// buffered TENSOR_LOAD_TO_LDS (28.8 KB per tile in flight -> latency-proof),
// compute reads LDS only (tid -> consecutive banks, conflict-free).
// ---------------------------------------------------------------------------
__global__ void __launch_bounds__(256)
hs_bottom_grouped(const float* __restrict__ x,      // [BATCH, NHID]
                  const float* __restrict__ Wb,     // [NCLS, NHID, NPC]
                  const float* __restrict__ bb,     // [NCLS, NPC]
                  const int*   __restrict__ counts, // [NCLS]
                  const int*   __restrict__ lists,  // [NCLS][BATCH]
                  float* __restrict__ botLog)       // [BATCH, NPC]
{
    __shared__ float wbuf[2][WROWS * NPC];  // 2 x 28.8 KB TDM targets
    __shared__ float xs[CHMAX][NHID];       // 32 KB sample rows
    __shared__ int   smp[CHMAX];

    const int c   = blockIdx.x;
    const int tid = threadIdx.x;
    const int n   = counts[c];
    if (n == 0) return;

    const bool wave0 = (tid < 32);          // TDM issued once, by wave 0 only
    const unsigned lbase0 = (unsigned)(uintptr_t)&wbuf[0][0];
    const unsigned lbase1 = (unsigned)(uintptr_t)&wbuf[1][0];
    const float* wslab = Wb + (size_t)c * (NHID * NPC);

    for (int base = 0; base < n; base += CHMAX) {
        const int m = min(CHMAX, n - base);

        if (tid < m) smp[tid] = lists[c * BATCH + base + tid];
        __syncthreads();

        for (int j = 0; j < m; ++j) {       // stage rows: one b128/thread/row
            const float4* xr = (const float4*)(x + (size_t)smp[j] * NHID);
            ((float4*)&xs[j][0])[tid] = xr[tid];
        }

        float acc[CHMAX];
#pragma unroll
        for (int j = 0; j < CHMAX; ++j) acc[j] = 0.0f;

        // prologue: fetch tile 0
        if (wave0) {
            tdm_load_w_tile(wslab, lbase0);
            __builtin_amdgcn_s_wait_tensorcnt(0);
        }
        __syncthreads();

        for (int ch = 0; ch < NCHUNK; ++ch) {
            const int cur = ch & 1;
            if (wave0 && (ch + 1) < NCHUNK)   // prefetch next tile into other buf
                tdm_load_w_tile(wslab + (size_t)(ch + 1) * WROWS * NPC,
                                cur ? lbase0 : lbase1);

            if (tid < NPC) {
                const float* wl = &wbuf[cur][0];
                const int    d0 = ch * WROWS;
#pragma unroll 4
                for (int dl = 0; dl < WROWS; ++dl) {
                    const float wv = wl[dl * NPC + tid];
#pragma unroll
                    for (int j = 0; j < CHMAX; ++j)
                        acc[j] = fmaf(xs[j][d0 + dl], wv, acc[j]);
                }
            }

            if (wave0) __builtin_amdgcn_s_wait_tensorcnt(0);
            __syncthreads();
        }

        if (tid < NPC) {
            const float bv = bb[(size_t)c * NPC + tid];
            for (int j = 0; j < m; ++j)
                botLog[(size_t)smp[j] * NPC + tid] = acc[j] + bv;
        }
        __syncthreads();                    // before next chunk reuses xs/smp
    }
}

// ---------------------------------------------------------------------------
// Kernel F: per-sample softmaxes + final product.
// ---------------------------------------------------------------------------
__global__ void __launch_bounds__(256)
hs_softmax_final(const int*   __restrict__ labels,
                 const float* __restrict__ topLog,  // [BATCH, NCLS]
                 const float* __restrict__ botLog,  // [BATCH, NPC]
                 float* __restrict__ out)           // [BATCH]
{
    __shared__ float red[256];
    __shared__ float sBot, sTop;

    const int b   = blockIdx.x;
    const int tid = threadIdx.x;
    const int lab = labels[b];
    const int ct  = lab / NPC;
    const int cb  = lab % NPC;

    const bool  act = (tid < NPC);
    const float bl  = act ? botLog[(size_t)b * NPC + tid] : -INFINITY;
    red[tid] = bl;
    __syncthreads();
    for (int s = 128; s > 0; s >>= 1) {
        if (tid < s) red[tid] = fmaxf(red[tid], red[tid + s]);
        __syncthreads();
    }
    const float bmax = red[0];
    __syncthreads();

    const float be = act ? __expf(bl - bmax) : 0.0f;
    red[tid] = be;
    __syncthreads();
    for (int s = 128; s > 0; s >>= 1) {
        if (tid < s) red[tid] += red[tid + s];
        __syncthreads();
    }
    const float bden = red[0];
    __syncthreads();
    if (tid == cb) sBot = be / bden;

    const bool  act2 = (tid < NCLS);
    const float tl   = act2 ? topLog[(size_t)b * NCLS + tid] : -INFINITY;
    red[tid] = tl;
    __syncthreads();
    for (int s = 128; s > 0; s >>= 1) {
        if (tid < s) red[tid] = fmaxf(red[tid], red[tid + s]);
        __syncthreads();
    }
    const float tmax = red[0];
    __syncthreads();

    const float te = act2 ? __expf(tl - tmax) : 0.0f;
    red[tid] = te;
    __syncthreads();
    for (int s = 128; s > 0; s >>= 1) {
        if (tid < s) red[tid] += red[tid + s];
        __syncthreads();
    }
    const float tden = red[0];
    if (tid == ct) sTop = te / tden;
    __syncthreads();

    if (tid == 0) out[b] = sTop * sBot;
}

// ---------------------------------------------------------------------------
// Launch. Inputs: {inputs, labels, top_W, top_b, bottom_W, bottom_b}.
// ws: topLog[1024*224]f | botLog[1024*225]f | counts[224]i | lists[224*1024]i
// ---------------------------------------------------------------------------
extern "C" void kernel_launch(void* const* d_in, const int* in_sizes, int n_in,
                              void* d_out, int out_size, void* d_ws, size_t ws_size,
                              hipStream_t stream)
{
    const float* inputs = (const float*)d_in[0];
    const int*   labels = (const int*)  d_in[1];
    const float* topW   = (const float*)d_in[2];
    const float* topB   = (const float*)d_in[3];
    const float* botW   = (const float*)d_in[4];
    const float* botB   = (const float*)d_in[5];
    float*       out    = (float*)d_out;

    char* ws = (char*)d_ws;
    float* topLog = (float*)ws;
    float* botLog = (float*)(ws + 917504);
    int*   counts = (int*)  (ws + 917504 + 921600);
    int*   lists  = (int*)  (ws + 917504 + 921600 + 896);

    dim3 gridA(BATCH / 16, NCLS / 16);
    hs_top_gemm_wmma<<<gridA, dim3(32), 0, stream>>>(inputs, topW, topB, topLog);

    hs_zero_counts<<<dim3(1),         dim3(256), 0, stream>>>(counts);
    hs_build_lists<<<dim3(BATCH/256), dim3(256), 0, stream>>>(labels, counts, lists);

    hs_bottom_grouped<<<dim3(NCLS), dim3(256), 0, stream>>>(
        inputs, botW, botB, counts, lists, botLog);

    hs_softmax_final<<<dim3(BATCH), dim3(256), 0, stream>>>(
        labels, topLog, botLog, out);
}